// SemiCRFModel_36507222016110
// MI455X (gfx1250) — compile-verified
//
#include <hip/hip_runtime.h>
#include <math.h>

#define L_SEQ   8192
#define F_DIM   26
#define SEG     100
#define HIDDEN  64
#define KPAD    56            // 54 agg dims padded to 56 (multiple of 4)
#define KP2     (KPAD / 2)    // 28 K-pairs
#define CHUNK   64
#define NCHUNK  (L_SEQ / CHUNK)   // 128
#define WPB     8             // waves per block (tiles per block)

typedef __attribute__((ext_vector_type(2))) float v2f;
typedef __attribute__((ext_vector_type(8))) float v8f;

// ---------------- prefix-sum stage (c1 = cumsum(x), c2 = cumsum(x*x)) -------

__global__ void pp_partial(const float* __restrict__ x, float* __restrict__ s1,
                           float* __restrict__ s2) {
  int b = blockIdx.x;
  int f = threadIdx.x;
  if (f >= F_DIM) return;
  float a1 = 0.f, a2 = 0.f;
  int base = b * CHUNK;
  for (int r = 0; r < CHUNK; ++r) {
    float v = x[(base + r) * F_DIM + f];
    a1 += v; a2 += v * v;
  }
  s1[b * F_DIM + f] = a1;
  s2[b * F_DIM + f] = a2;
}

__global__ void pp_scan(float* __restrict__ s1, float* __restrict__ s2) {
  int f = threadIdx.x;
  if (f >= F_DIM) return;
  float r1 = 0.f, r2 = 0.f;
  for (int b = 0; b < NCHUNK; ++b) {
    float t1 = s1[b * F_DIM + f]; s1[b * F_DIM + f] = r1; r1 += t1;
    float t2 = s2[b * F_DIM + f]; s2[b * F_DIM + f] = r2; r2 += t2;
  }
}

__global__ void pp_write(const float* __restrict__ x, const float* __restrict__ s1,
                         const float* __restrict__ s2, float* __restrict__ c1,
                         float* __restrict__ c2) {
  int b = blockIdx.x;
  int f = threadIdx.x;
  if (f >= F_DIM) return;
  float r1 = s1[b * F_DIM + f];
  float r2 = s2[b * F_DIM + f];
  if (b == 0) { c1[f] = 0.f; c2[f] = 0.f; }
  int base = b * CHUNK;
  for (int r = 0; r < CHUNK; ++r) {
    float v = x[(base + r) * F_DIM + f];
    r1 += v; r2 += v * v;
    c1[(base + r + 1) * F_DIM + f] = r1;
    c2[(base + r + 1) * F_DIM + f] = r2;
  }
}

// ---------------- main span-score kernel: agg -> WMMA MLP -------------------

__launch_bounds__(32 * WPB)
__global__ void span_score(const float* __restrict__ c1, const float* __restrict__ c2,
                           const float* __restrict__ pW1, const float* __restrict__ pb1,
                           const float* __restrict__ pW2, const float* __restrict__ pb2,
                           float* __restrict__ out) {
  // W1 staged K-pair-interleaved: W1P[kp*HIDDEN + n] = (W1[2kp][n], W1[2kp+1][n])
  // so every WMMA B-fragment is one aligned ds_load_b64.
  __shared__ v2f   W1P[KP2 * HIDDEN];         // 28*64 float2 = 14 KB
  __shared__ float aggL[WPB][16 * KPAD];      // per-wave 16x56 A-tiles
  __shared__ float scoreL[WPB * 16];

  const int tid  = threadIdx.x;
  const int wave = tid >> 5;
  const int lane = tid & 31;
  const int m    = lane & 15;   // span-in-tile / N-column
  const int h    = lane >> 4;   // lane half

  // Stage zero-padded, pair-interleaved W1 into LDS (shared by all waves).
  for (int idx = tid; idx < KP2 * HIDDEN; idx += 32 * WPB) {
    int kp = idx >> 6;         // / HIDDEN
    int n  = idx & 63;         // % HIDDEN
    v2f w;
    if (kp < 27) {             // 2kp, 2kp+1 both < 54
      w.x = pW1[(2 * kp)     * HIDDEN + n];
      w.y = pW1[(2 * kp + 1) * HIDDEN + n];
    } else {                   // pad pair (K = 54, 55)
      w.x = 0.f; w.y = 0.f;
    }
    W1P[idx] = w;
  }

  // Identify this lane's span.
  const int t0 = (blockIdx.x * WPB + wave) * 16;
  const int t  = t0 + m;
  const int i  = t / SEG;
  const int l  = t % SEG + 1;
  const int j  = i + l;
  const bool valid = (j <= L_SEQ);
  const int jc = valid ? j : L_SEQ;
  const float lf   = (float)l;
  const float invl = 1.0f / lf;

  // Build agg row for span m. Uniform lane split: each lane computes the
  // mean AND variance for 13 features (g = 13h + q), reusing the c1 diff.
  const float* c1i = c1 + (size_t)i  * F_DIM;
  const float* c1j = c1 + (size_t)jc * F_DIM;
  const float* c2i = c2 + (size_t)i  * F_DIM;
  const float* c2j = c2 + (size_t)jc * F_DIM;
  float* arow = &aggL[wave][m * KPAD];
  const int gbase = 13 * h;

  float meanv[13], varv[13];
#pragma unroll
  for (int q = 0; q < 13; ++q) {
    int g = gbase + q;
    float d1 = c1j[g] - c1i[g];
    float d2 = c2j[g] - c2i[g];
    float mean = d1 * invl;
    meanv[q] = mean;
    varv[q]  = d2 * invl - mean * mean;
  }
#pragma unroll
  for (int q = 0; q < 13; ++q) arow[gbase + q] = meanv[q];
#pragma unroll
  for (int q = 0; q < 13; ++q) arow[F_DIM + gbase + q] = varv[q];
  if (h == 0) {
    arow[2 * F_DIM]     = lf * (1.0f / 500.0f);                 // len feature
    arow[2 * F_DIM + 1] = __logf(lf + 1e-6f) * (1.0f / 6.0f);   // loglen feature
  } else {
    arow[2 * F_DIM + 2] = 0.f;                                  // K padding
    arow[2 * F_DIM + 3] = 0.f;
  }
  __syncthreads();

  // GEMM: D(16x64) = A(16x56) x W1(56x64), f32 WMMA, K in steps of 4.
  // Lane-half h owns K-pair (4k+2h, 4k+2h+1) of each step -> pair index 2k+h.
  v8f acc0 = {}, acc1 = {}, acc2 = {}, acc3 = {};
  const float* aw = arow;
#pragma unroll
  for (int k = 0; k < KPAD / 4; ++k) {
    const int kp = 2 * k + h;
    v2f a  = *(const v2f*)(aw + 2 * kp);
    v2f b0 = W1P[kp * HIDDEN +  0 + m];
    v2f b1 = W1P[kp * HIDDEN + 16 + m];
    v2f b2 = W1P[kp * HIDDEN + 32 + m];
    v2f b3 = W1P[kp * HIDDEN + 48 + m];
    acc0 = __builtin_amdgcn_wmma_f32_16x16x4_f32(false, a, false, b0, (short)0, acc0, false, false);
    acc1 = __builtin_amdgcn_wmma_f32_16x16x4_f32(false, a, false, b1, (short)0, acc1, false, false);
    acc2 = __builtin_amdgcn_wmma_f32_16x16x4_f32(false, a, false, b2, (short)0, acc2, false, false);
    acc3 = __builtin_amdgcn_wmma_f32_16x16x4_f32(false, a, false, b3, (short)0, acc3, false, false);
  }

  // bias + ReLU + dot with W2, then reduce over the 16 N-lanes of each half.
  const float bb0 = pb1[ 0 + m], bb1 = pb1[16 + m], bb2 = pb1[32 + m], bb3 = pb1[48 + m];
  const float w20 = pW2[ 0 + m], w21 = pW2[16 + m], w22 = pW2[32 + m], w23 = pW2[48 + m];
  float sc[8];
#pragma unroll
  for (int r = 0; r < 8; ++r) {
    float p = fmaxf(acc0[r] + bb0, 0.f) * w20
            + fmaxf(acc1[r] + bb1, 0.f) * w21
            + fmaxf(acc2[r] + bb2, 0.f) * w22
            + fmaxf(acc3[r] + bb3, 0.f) * w23;
    p += __shfl_xor(p, 1, 32);
    p += __shfl_xor(p, 2, 32);
    p += __shfl_xor(p, 4, 32);
    p += __shfl_xor(p, 8, 32);
    sc[r] = p;       // row r (lanes 0-15) / row r+8 (lanes 16-31)
  }
  if (m == 0) {
#pragma unroll
    for (int r = 0; r < 8; ++r) scoreL[wave * 16 + h * 8 + r] = sc[r];
  }
  __syncthreads();

  if (lane < 16) {
    float s = scoreL[wave * 16 + m] + pb2[0];
    out[t] = valid ? s : 0.f;
  }
}

// ---------------- launcher --------------------------------------------------

extern "C" void kernel_launch(void* const* d_in, const int* in_sizes, int n_in,
                              void* d_out, int out_size, void* d_ws, size_t ws_size,
                              hipStream_t stream) {
  const float* res = (const float*)d_in[0];   // (8192, 26)
  const float* W1  = (const float*)d_in[1];   // (54, 64)
  const float* b1  = (const float*)d_in[2];   // (64,)
  const float* W2  = (const float*)d_in[3];   // (64, 1)
  const float* b2  = (const float*)d_in[4];   // (1,)
  float* out = (float*)d_out;                 // (8192, 100)

  float* c1 = (float*)d_ws;                         // (L+1, F)
  float* c2 = c1 + (size_t)(L_SEQ + 1) * F_DIM;     // (L+1, F)
  float* s1 = c2 + (size_t)(L_SEQ + 1) * F_DIM;     // (NCHUNK, F)
  float* s2 = s1 + (size_t)NCHUNK * F_DIM;          // (NCHUNK, F)

  pp_partial<<<NCHUNK, 32, 0, stream>>>(res, s1, s2);
  pp_scan<<<1, 32, 0, stream>>>(s1, s2);
  pp_write<<<NCHUNK, 32, 0, stream>>>(res, s1, s2, c1, c2);

  const int total_tiles = (L_SEQ * SEG) / 16;       // 51200
  const int blocks = total_tiles / WPB;             // 6400
  span_score<<<blocks, 32 * WPB, 0, stream>>>(c1, c2, W1, b1, W2, b2, out);
}